// GraphSAGE_29308856828029
// MI455X (gfx1250) — compile-verified
//
#include <hip/hip_runtime.h>
#include <hip/hip_bf16.h>

typedef float v2f __attribute__((ext_vector_type(2)));
typedef float v8f __attribute__((ext_vector_type(8)));

#define D 128   // feature dim for all layers

// ---------------------------------------------------------------------------
// Zero a float buffer
// ---------------------------------------------------------------------------
__global__ void zero_kernel(float* __restrict__ p, long long n) {
    long long i = (long long)blockIdx.x * blockDim.x + threadIdx.x;
    if (i < n) p[i] = 0.0f;
}

// ---------------------------------------------------------------------------
// deg[dst[e]] += 1 for every edge
// ---------------------------------------------------------------------------
__global__ void degree_kernel(const int* __restrict__ dst, float* __restrict__ deg, int E) {
    int e = blockIdx.x * blockDim.x + threadIdx.x;
    if (e < E) atomicAdd(&deg[dst[e]], 1.0f);
}

// ---------------------------------------------------------------------------
// Scatter-add: one wave (32 lanes) per edge, lane handles 4 consecutive floats.
// Source-row read is a fully coalesced 512B float4 burst; atomics land in L2
// (whole agg buffer is 25.6MB, fits in the 192MB L2).
// ---------------------------------------------------------------------------
__global__ __launch_bounds__(256) void scatter_kernel(
    const float* __restrict__ feat, const int* __restrict__ src,
    const int* __restrict__ dst, float* __restrict__ agg, int E)
{
    int wave = (int)((blockIdx.x * (long long)blockDim.x + threadIdx.x) >> 5);
    int lane = threadIdx.x & 31;
    if (wave >= E) return;
    int s = src[wave];
    int d = dst[wave];
    const float4 v = *(const float4*)(feat + (size_t)s * D + lane * 4);
    float* p = agg + (size_t)d * D + lane * 4;
    atomicAdd(p + 0, v.x);
    atomicAdd(p + 1, v.y);
    atomicAdd(p + 2, v.z);
    atomicAdd(p + 3, v.w);
}

// ---------------------------------------------------------------------------
// Fused SAGE linear: out = relu?( (agg/max(deg,1)) @ Wl + hin @ Wr + b )
// Block = 256 threads = 8 waves, covers 16 output rows x 128 output cols.
// agg tile (scaled) and hin tile staged in LDS; each wave owns one 16x16
// column tile and runs v_wmma_f32_16x16x4_f32 over K=128.
// ---------------------------------------------------------------------------
__global__ __launch_bounds__(256) void sage_gemm_kernel(
    const float* __restrict__ agg, const float* __restrict__ deg,
    const float* __restrict__ hin, const float* __restrict__ Wl,
    const float* __restrict__ Wr, const float* __restrict__ bias,
    float* __restrict__ out, int relu)
{
    __shared__ float sA[16 * D];   // degree-normalized agg rows
    __shared__ float sH[16 * D];   // self feature rows

    const int rowBase = blockIdx.x * 16;
    const int tid = threadIdx.x;

    // Cooperative load: 2 x 2048 floats, float4 per thread per iteration.
    for (int i = tid; i < 16 * D / 4; i += 256) {
        int idx = i * 4;
        int r = idx >> 7;           // row within tile (idx / 128)
        int c = idx & (D - 1);      // col within row
        float inv = 1.0f / fmaxf(deg[rowBase + r], 1.0f);
        float4 va = *(const float4*)(agg + (size_t)(rowBase + r) * D + c);
        va.x *= inv; va.y *= inv; va.z *= inv; va.w *= inv;
        *(float4*)(sA + idx) = va;
        *(float4*)(sH + idx) = *(const float4*)(hin + (size_t)(rowBase + r) * D + c);
    }
    __syncthreads();

    const int wave = tid >> 5;          // 0..7 -> 16-wide column tile
    const int lane = tid & 31;
    const int m = lane & 15;            // A-row / D-column index within tile
    const int col = wave * 16 + m;      // global output column
    const int khalf = (lane >> 4) * 2;  // lanes 0-15: K+0/K+1, lanes 16-31: K+2/K+3

    v8f acc;
    const float bv = bias[col];
#pragma unroll
    for (int i = 0; i < 8; ++i) acc[i] = bv;

#pragma unroll 4
    for (int k = 0; k < D; k += 4) {
        v2f a, b;
        // A fragment from LDS (16x4 f32 layout, ISA 7.12.2)
        a.x = sA[m * D + k + khalf];
        a.y = sA[m * D + k + khalf + 1];
        // B fragment: Wl is row-major [K, 128]
        b.x = Wl[(size_t)(k + khalf) * D + col];
        b.y = Wl[(size_t)(k + khalf + 1) * D + col];
        acc = __builtin_amdgcn_wmma_f32_16x16x4_f32(
            false, a, false, b, (short)0, acc, false, false);

        a.x = sH[m * D + k + khalf];
        a.y = sH[m * D + k + khalf + 1];
        b.x = Wr[(size_t)(k + khalf) * D + col];
        b.y = Wr[(size_t)(k + khalf + 1) * D + col];
        acc = __builtin_amdgcn_wmma_f32_16x16x4_f32(
            false, a, false, b, (short)0, acc, false, false);
    }

    // D layout: lanes 0-15 hold rows 0..7, lanes 16-31 hold rows 8..15.
    const int rOff = (lane >> 4) * 8;
#pragma unroll
    for (int r = 0; r < 8; ++r) {
        float v = acc[r];
        if (relu) v = fmaxf(v, 0.0f);
        out[(size_t)(rowBase + rOff + r) * D + col] = v;
    }
}

// ---------------------------------------------------------------------------
extern "C" void kernel_launch(void* const* d_in, const int* in_sizes, int n_in,
                              void* d_out, int out_size, void* d_ws, size_t ws_size,
                              hipStream_t stream) {
    const float* x   = (const float*)d_in[0];
    const int*  edge = (const int*) d_in[1];   // [2, E]: row0=src, row1=dst
    const float* W1l = (const float*)d_in[2];
    const float* b1  = (const float*)d_in[3];
    const float* W1r = (const float*)d_in[4];
    const float* W2l = (const float*)d_in[5];
    const float* b2  = (const float*)d_in[6];
    const float* W2r = (const float*)d_in[7];

    const int N = in_sizes[0] / D;     // 50000
    const int E = in_sizes[1] / 2;     // 800000

    // Workspace layout: [agg: N*D][deg: N][h: N*D]
    float* agg = (float*)d_ws;
    float* deg = agg + (size_t)N * D;
    float* h   = deg + N;

    const int* src = edge;
    const int* dst = edge + E;
    float* out = (float*)d_out;

    const long long aggN = (long long)N * D;
    const int scatterBlocks = (E + 7) / 8;       // 8 waves(=edges) per 256-thr block
    const int rowTiles = N / 16;                 // 50000 / 16 = 3125 exact

    // ---- Layer 1 ----
    zero_kernel<<<(unsigned)((aggN + N + 255) / 256), 256, 0, stream>>>(agg, aggN + N);
    degree_kernel<<<(E + 255) / 256, 256, 0, stream>>>(dst, deg, E);
    scatter_kernel<<<scatterBlocks, 256, 0, stream>>>(x, src, dst, agg, E);
    sage_gemm_kernel<<<rowTiles, 256, 0, stream>>>(agg, deg, x, W1l, W1r, b1, h, 1);

    // ---- Layer 2 ----
    zero_kernel<<<(unsigned)((aggN + 255) / 256), 256, 0, stream>>>(agg, aggN);
    scatter_kernel<<<scatterBlocks, 256, 0, stream>>>(h, src, dst, agg, E);
    sage_gemm_kernel<<<rowTiles, 256, 0, stream>>>(agg, deg, h, W2l, W2r, b2, out, 0);
}